// BGAN_74345883894232
// MI455X (gfx1250) — compile-verified
//
#include <hip/hip_runtime.h>
#include <hip/hip_bf16.h>

// ---------------- problem constants ----------------
#define NN     50000
#define IND    128
#define OUTD   256
#define KNB    10          // K neighbors
#define EE     (NN * 32)   // edges
#define NCLS   10
#define ROWD   1161        // 9*129 row-conv outputs per node
#define LOCALD 1417        // ROWD + OUTD
#define OUTP   1424        // LOCALD padded to multiple of 16 (zero pad)
#define EPSV   1e-5f
#define SLOPE  0.01f
#define MTILES (NN / 16)   // 3125

typedef __attribute__((ext_vector_type(2))) float v2f;
typedef __attribute__((ext_vector_type(8))) float v8f;

__device__ __forceinline__ void atomicAddF(float* p, float v) {
  __hip_atomic_fetch_add(p, v, __ATOMIC_RELAXED, __HIP_MEMORY_SCOPE_AGENT);
}

__device__ __forceinline__ float wave_sum(float v) {
#pragma unroll
  for (int m = 16; m; m >>= 1) v += __shfl_xor(v, m, 32);
  return v;
}

// ============ K0: zero-padded copy of localw (1424 x 256, L2-resident) ============
__global__ __launch_bounds__(256) void k_padw(const float* __restrict__ localw,
                                              float* __restrict__ lwpad) {
  const int i = blockIdx.x * 256 + threadIdx.x;     // over OUTP*OUTD
  if (i >= OUTP * OUTD) return;
  const int j = i >> 8;                             // row (OUTD == 256)
  lwpad[i] = (j < LOCALD) ? localw[i] : 0.f;
}

// ================= K1: z = h @ fc_w^T  (WMMA f32 16x16x4) =================
__global__ __launch_bounds__(256) void k_gemm_z(const float* __restrict__ h,
                                                const float* __restrict__ fc_w,
                                                float* __restrict__ z) {
  const int lane = threadIdx.x & 31, wv = threadIdx.x >> 5;
  const long tile = (long)blockIdx.x * 8 + wv;          // 50000 tiles
  const int mt = (int)(tile >> 4), nt = (int)(tile & 15);
  if (mt >= MTILES) return;
  const int row   = mt * 16 + (lane & 15);
  const int col   = nt * 16 + (lane & 15);
  const int khalf = (lane >> 4) * 2;                     // 0 or 2
  const float* __restrict__ ap = h    + (long)row * IND;
  const float* __restrict__ bp = fc_w + (long)col * IND; // B[k][n] = fc_w[n][k]
  v8f acc = {};
#pragma unroll 4
  for (int k0 = 0; k0 < IND; k0 += 4) {
    v2f a = *(const v2f*)(ap + k0 + khalf);
    v2f b = *(const v2f*)(bp + k0 + khalf);
    acc = __builtin_amdgcn_wmma_f32_16x16x4_f32(false, a, false, b,
                                                (short)0, acc, false, false);
  }
  const int rbase = mt * 16 + (lane >> 4) * 8;
  const int c     = nt * 16 + (lane & 15);
#pragma unroll
  for (int r = 0; r < 8; ++r)
    z[(long)(rbase + r) * OUTD + c] = acc[r];
}

// ================= K2a: per-node attention dots =================
__global__ __launch_bounds__(256) void k_dots(const float* __restrict__ z,
                                              const float* __restrict__ attn_w,
                                              float* __restrict__ dsrc,
                                              float* __restrict__ ddst) {
  const int lane = threadIdx.x & 31, wv = threadIdx.x >> 5;
  const int n = blockIdx.x * 8 + wv;
  if (n >= NN) return;
  float s0 = 0.f, s1 = 0.f;
  for (int d = lane; d < OUTD; d += 32) {
    float zv = z[(long)n * OUTD + d];
    s0 = fmaf(zv, attn_w[d], s0);
    s1 = fmaf(zv, attn_w[OUTD + d], s1);
  }
  s0 = wave_sum(s0); s1 = wave_sum(s1);
  if (lane == 0) { dsrc[n] = s0; ddst[n] = s1; }
}

// ================= K2b: leaky-relu + softmax over K = alpha =================
__global__ __launch_bounds__(256) void k_alpha(const int* __restrict__ nidx,
                                               const float* __restrict__ dsrc,
                                               const float* __restrict__ ddst,
                                               float* __restrict__ alpha) {
  const int n = blockIdx.x * blockDim.x + threadIdx.x;
  if (n >= NN) return;
  const float dd = ddst[n];
  float e[KNB], mx = -3.4e38f;
#pragma unroll
  for (int k = 0; k < KNB; ++k) {
    float v = dsrc[nidx[n * KNB + k]] + dd;
    v = v > 0.f ? v : SLOPE * v;
    e[k] = v; mx = fmaxf(mx, v);
  }
  float s = 0.f;
#pragma unroll
  for (int k = 0; k < KNB; ++k) { e[k] = __expf(e[k] - mx); s += e[k]; }
  const float inv = 1.f / s;
#pragma unroll
  for (int k = 0; k < KNB; ++k) alpha[n * KNB + k] = e[k] * inv;
}

// ========== K3: per-node feats in LDS -> row conv + col conv + BN sums ==========
// Writes raw (pre-BN) conv outputs into outbuf[n*OUTP + {0..1160 row, 1161..1416 col}]
__global__ __launch_bounds__(256) void k_rowcol(const float* __restrict__ z,
                                                const int* __restrict__ nidx,
                                                const float* __restrict__ alpha,
                                                const float* __restrict__ convrow_w,
                                                const float* __restrict__ convrow_b,
                                                const float* __restrict__ convcol_w,
                                                const float* __restrict__ convcol_b,
                                                float* __restrict__ outbuf,
                                                float* __restrict__ accum) {
  __shared__ float fe[KNB * OUTD];   // 10 KB: alpha-weighted gathered features
  __shared__ float cw[2 * IND];
  __shared__ float av[KNB];
  __shared__ int   jv[KNB];
  __shared__ float cwc[KNB];
  __shared__ float rr[8][4];
  const int n = blockIdx.x, tid = threadIdx.x;
  if (tid < KNB)       { av[tid] = alpha[n * KNB + tid]; jv[tid] = nidx[n * KNB + tid]; }
  if (tid >= 32 && tid < 32 + KNB) cwc[tid - 32] = convcol_w[tid - 32];
  cw[tid] = convrow_w[tid];          // 256 threads == 2*IND
  __syncthreads();
#pragma unroll
  for (int k = 0; k < KNB; ++k)
    fe[k * OUTD + tid] = av[k] * z[(long)jv[k] * OUTD + tid];
  __syncthreads();
  const float rb = convrow_b[0], cb = convcol_b[0];
  float* __restrict__ op = outbuf + (long)n * OUTP;
  // column conv (einsum nkd,k->nd)
  float cval = cb;
#pragma unroll
  for (int k = 0; k < KNB; ++k) cval = fmaf(fe[k * OUTD + tid], cwc[k], cval);
  op[ROWD + tid] = cval;
  float s_c = cval, ss_c = cval * cval;
  // row conv: 9 x 129 VALID correlation with 2x128 filter
  float s_r = 0.f, ss_r = 0.f;
  for (int j = tid; j < ROWD; j += 256) {
    const int kk = j / 129, w = j % 129;
    const float* f0 = fe + kk * OUTD + w;
    const float* f1 = f0 + OUTD;
    float r = rb;
#pragma unroll 4
    for (int dw = 0; dw < IND; ++dw)
      r = fmaf(f0[dw], cw[dw], fmaf(f1[dw], cw[IND + dw], r));
    op[j] = r;
    s_r += r; ss_r += r * r;
  }
  s_r = wave_sum(s_r); ss_r = wave_sum(ss_r);
  s_c = wave_sum(s_c); ss_c = wave_sum(ss_c);
  const int lane = tid & 31, wv = tid >> 5;
  if (lane == 0) { rr[wv][0] = s_r; rr[wv][1] = ss_r; rr[wv][2] = s_c; rr[wv][3] = ss_c; }
  __syncthreads();
  if (tid == 0) {
    float a0 = 0, a1 = 0, a2 = 0, a3 = 0;
#pragma unroll
    for (int w = 0; w < 8; ++w) { a0 += rr[w][0]; a1 += rr[w][1]; a2 += rr[w][2]; a3 += rr[w][3]; }
    atomicAddF(accum + 0, a0); atomicAddF(accum + 1, a1);
    atomicAddF(accum + 2, a2); atomicAddF(accum + 3, a3);
  }
}

// ================= K4: fold BN stats into scale/shift =================
__global__ void k_bnparams(const float* __restrict__ accum,
                           const float* __restrict__ g_r, const float* __restrict__ b_r,
                           const float* __restrict__ g_c, const float* __restrict__ b_c,
                           float* __restrict__ params) {
  if (threadIdx.x == 0 && blockIdx.x == 0) {
    const float Mr = (float)NN * (float)ROWD, Mc = (float)NN * (float)OUTD;
    float mr = accum[0] / Mr, vr = accum[1] / Mr - mr * mr;
    float rs = g_r[0] * rsqrtf(vr + EPSV);
    float mc = accum[2] / Mc, vc = accum[3] / Mc - mc * mc;
    float cs = g_c[0] * rsqrtf(vc + EPSV);
    params[0] = rs; params[1] = b_r[0] - mr * rs;
    params[2] = cs; params[3] = b_c[0] - mc * cs;
  }
}

// ====== K4b: apply BN affine + ReLU in place, zero the pad tail ======
__global__ __launch_bounds__(256) void k_bnapply(float* __restrict__ outbuf,
                                                 const float* __restrict__ params) {
  const int n = blockIdx.x, tid = threadIdx.x;
  const float rs = params[0], rsh = params[1], cs = params[2], csh = params[3];
  float* __restrict__ p = outbuf + (long)n * OUTP;
  for (int j = tid; j < OUTP; j += 256) {
    float v;
    if (j < ROWD)        v = fmaxf(fmaf(p[j], rs, rsh), 0.f);
    else if (j < LOCALD) v = fmaxf(fmaf(p[j], cs, csh), 0.f);
    else                 v = 0.f;                    // pad: contributes nothing
    p[j] = v;
  }
}

// ================= K6: degrees via f32 atomics =================
__global__ __launch_bounds__(256) void k_deg(const int* __restrict__ src,
                                             const int* __restrict__ dst,
                                             float* __restrict__ dego,
                                             float* __restrict__ degi) {
  const int e = blockIdx.x * blockDim.x + threadIdx.x;
  if (e >= EE) return;
  atomicAddF(dego + src[e], 1.f);
  atomicAddF(degi + dst[e], 1.f);
}

// ================= K7: feat = (h @ gc_w) * rsqrt(max(deg_out,1)) ==========
__global__ __launch_bounds__(256) void k_featgc(const float* __restrict__ h,
                                                const float* __restrict__ gc_w,
                                                const float* __restrict__ dego,
                                                float* __restrict__ featgc) {
  const int lane = threadIdx.x & 31, wv = threadIdx.x >> 5;
  const int n = blockIdx.x * 8 + wv;
  if (n >= NN) return;
  float s = 0.f;
  for (int d = lane; d < IND; d += 32) s = fmaf(h[(long)n * IND + d], gc_w[d], s);
  s = wave_sum(s);
  if (lane == 0) featgc[n] = s * rsqrtf(fmaxf(dego[n], 1.f));
}

// ================= K8: agg = segment_sum(feat[src], dst) =================
__global__ __launch_bounds__(256) void k_agg(const int* __restrict__ src,
                                             const int* __restrict__ dst,
                                             const float* __restrict__ featgc,
                                             float* __restrict__ agg) {
  const int e = blockIdx.x * blockDim.x + threadIdx.x;
  if (e >= EE) return;
  atomicAddF(agg + dst[e], featgc[src[e]]);
}

// ================= K9a: scores + per-block max =================
__global__ __launch_bounds__(256) void k_scores(const float* __restrict__ agg,
                                                const float* __restrict__ degi,
                                                const float* __restrict__ gc_b,
                                                float* __restrict__ scores,
                                                float* __restrict__ partial) {
  __shared__ float red[256];
  const int n = blockIdx.x * 256 + threadIdx.x;
  float sc = -3.4e38f;
  if (n < NN) {
    sc = agg[n] * rsqrtf(fmaxf(degi[n], 1.f)) + gc_b[0];
    scores[n] = sc;
  }
  red[threadIdx.x] = sc; __syncthreads();
  for (int s = 128; s; s >>= 1) {
    if (threadIdx.x < s) red[threadIdx.x] = fmaxf(red[threadIdx.x], red[threadIdx.x + s]);
    __syncthreads();
  }
  if (threadIdx.x == 0) partial[blockIdx.x] = red[0];
}

// ================= K9b: final max -> params[4] =================
__global__ __launch_bounds__(256) void k_maxfin(const float* __restrict__ partial,
                                                int nb, float* __restrict__ params) {
  __shared__ float red[256];
  float v = -3.4e38f;
  for (int i = threadIdx.x; i < nb; i += 256) v = fmaxf(v, partial[i]);
  red[threadIdx.x] = v; __syncthreads();
  for (int s = 128; s; s >>= 1) {
    if (threadIdx.x < s) red[threadIdx.x] = fmaxf(red[threadIdx.x], red[threadIdx.x + s]);
    __syncthreads();
  }
  if (threadIdx.x == 0) params[4] = red[0];
}

// ================= K9c: sumexp -> params[5] =================
__global__ __launch_bounds__(256) void k_sumexp(const float* __restrict__ scores,
                                                float* __restrict__ params) {
  __shared__ float red[256];
  const int n = blockIdx.x * 256 + threadIdx.x;
  const float smax = params[4];
  float v = (n < NN) ? __expf(scores[n] - smax) : 0.f;
  red[threadIdx.x] = v; __syncthreads();
  for (int s = 128; s; s >>= 1) {
    if (threadIdx.x < s) red[threadIdx.x] += red[threadIdx.x + s];
    __syncthreads();
  }
  if (threadIdx.x == 0) atomicAddF(params + 5, red[0]);
}

// ==== K5: relu-outputs GEMM: OUT@lwpad + h@h_proj^T, fused weighted reduce ====
// OUT is pre-activated & zero-padded -> uniform guard-free K loop of 356 steps.
__global__ __launch_bounds__(256) void k_update(const float* __restrict__ outbuf,
                                                const float* __restrict__ lwpad,
                                                const float* __restrict__ h,
                                                const float* __restrict__ h_proj_w,
                                                const float* __restrict__ params,
                                                const float* __restrict__ scores,
                                                float* __restrict__ hg) {
  const int lane = threadIdx.x & 31, wv = threadIdx.x >> 5;
  const long tile = (long)blockIdx.x * 8 + wv;
  const int mt = (int)(tile >> 4), nt = (int)(tile & 15);
  if (mt >= MTILES) return;
  const float smax = params[4];
  const int mrow  = mt * 16 + (lane & 15);
  const int col   = nt * 16 + (lane & 15);
  const int khalf = (lane >> 4) * 2;
  const float* __restrict__ ap = outbuf + (long)mrow * OUTP;
  const float* __restrict__ bp = lwpad + col;
  v8f acc = {};
  // --- OUT @ localw (padded): 356 uniform k-steps, aligned b64 A loads ---
#pragma unroll 2
  for (int k0 = 0; k0 < OUTP; k0 += 4) {
    const int k = k0 + khalf;
    v2f a = *(const v2f*)(ap + k);
    v2f b;
    b.x = bp[(long)k * OUTD];
    b.y = bp[(long)(k + 1) * OUTD];
    if (k0 + 8 < OUTP)
      __builtin_prefetch((const void*)(bp + (long)(k + 8) * OUTD), 0, 1);
    acc = __builtin_amdgcn_wmma_f32_16x16x4_f32(false, a, false, b,
                                                (short)0, acc, false, false);
  }
  // --- + h @ h_proj_w^T, K = 128 ---
  const float* __restrict__ hp = h + (long)mrow * IND;
  const float* __restrict__ pp = h_proj_w + (long)col * IND;
#pragma unroll 4
  for (int k0 = 0; k0 < IND; k0 += 4) {
    v2f a = *(const v2f*)(hp + k0 + khalf);
    v2f b = *(const v2f*)(pp + k0 + khalf);
    acc = __builtin_amdgcn_wmma_f32_16x16x4_f32(false, a, false, b,
                                                (short)0, acc, false, false);
  }
  // --- epilogue: relu, softmax weight, column-reduce into hg (unnormalized) ---
  const int rbase = mt * 16 + (lane >> 4) * 8;
  float colsum = 0.f;
#pragma unroll
  for (int r = 0; r < 8; ++r) {
    float u = fmaxf(acc[r], 0.f);
    float w = __expf(scores[rbase + r] - smax);
    colsum = fmaf(w, u, colsum);
  }
  colsum += __shfl_xor(colsum, 16, 32);     // join rows 0-7 with rows 8-15 (same col)
  if (lane < 16) atomicAddF(hg + nt * 16 + lane, colsum);
}

// ================= K10: logits = (hg/(N*sumexp)) @ cls_w^T + cls_b ==========
__global__ __launch_bounds__(320) void k_logits(const float* __restrict__ hg,
                                                const float* __restrict__ params,
                                                const float* __restrict__ cls_w,
                                                const float* __restrict__ cls_b,
                                                float* __restrict__ out) {
  const int lane = threadIdx.x & 31, wv = threadIdx.x >> 5;  // 10 waves, one per class
  if (wv >= NCLS) return;
  const float norm = 1.f / ((float)NN * params[5]);
  float s = 0.f;
  for (int d = lane; d < OUTD; d += 32)
    s = fmaf(hg[d] * norm, cls_w[wv * OUTD + d], s);
  s = wave_sum(s);
  if (lane == 0) out[wv] = s + cls_b[wv];
}

// =========================== host launcher ===========================
extern "C" void kernel_launch(void* const* d_in, const int* in_sizes, int n_in,
                              void* d_out, int out_size, void* d_ws, size_t ws_size,
                              hipStream_t stream) {
  (void)in_sizes; (void)n_in; (void)out_size; (void)ws_size;
  const float* h         = (const float*)d_in[0];
  const int*   nidx      = (const int*)  d_in[1];
  const int*   src       = (const int*)  d_in[2];
  const int*   dst       = (const int*)  d_in[3];
  const float* fc_w      = (const float*)d_in[4];
  const float* attn_w    = (const float*)d_in[5];
  const float* convrow_w = (const float*)d_in[6];
  const float* convrow_b = (const float*)d_in[7];
  const float* bnrow_g   = (const float*)d_in[8];
  const float* bnrow_b   = (const float*)d_in[9];
  const float* convcol_w = (const float*)d_in[10];
  const float* convcol_b = (const float*)d_in[11];
  const float* bncol_g   = (const float*)d_in[12];
  const float* bncol_b   = (const float*)d_in[13];
  const float* localw    = (const float*)d_in[14];
  const float* h_proj_w  = (const float*)d_in[15];
  const float* gc_w      = (const float*)d_in[16];
  const float* gc_b      = (const float*)d_in[17];
  const float* cls_w     = (const float*)d_in[18];
  const float* cls_b     = (const float*)d_in[19];

  float* ws = (float*)d_ws;
  size_t off = 0;
  float* z       = ws + off; off += (size_t)NN * OUTD;
  float* dsrc    = ws + off; off += NN;
  float* ddst    = ws + off; off += NN;
  float* alpha   = ws + off; off += (size_t)NN * KNB;
  float* outbuf  = ws + off; off += (size_t)NN * OUTP;   // row|col (padded, 16-aligned)
  float* lwpad   = ws + off; off += (size_t)OUTP * OUTD; // zero-padded localw
  float* featgc  = ws + off; off += NN;
  float* scores  = ws + off; off += NN;
  float* partial = ws + off; off += 256;
  float* zregion = ws + off;                 // everything below is zero-inited
  float* dego    = ws + off; off += NN;
  float* degi    = ws + off; off += NN;
  float* agg     = ws + off; off += NN;
  float* accum   = ws + off; off += 8;
  float* params  = ws + off; off += 8;
  float* hg      = ws + off; off += 256;
  const size_t zbytes = (size_t)(3 * NN + 8 + 8 + 256) * sizeof(float);
  hipMemsetAsync(zregion, 0, zbytes, stream);

  const int edgeBlocks = (EE + 255) / 256;       // 6250
  const int nodeBlocks = (NN + 255) / 256;       // 196
  const int tileBlocks = NN / 8;                 // 6250 (8 wave-tiles / block)
  const int padwBlocks = (OUTP * OUTD + 255) / 256;

  k_padw   <<<padwBlocks, 256, 0, stream>>>(localw, lwpad);
  k_gemm_z <<<tileBlocks, 256, 0, stream>>>(h, fc_w, z);
  k_dots   <<<tileBlocks, 256, 0, stream>>>(z, attn_w, dsrc, ddst);
  k_alpha  <<<nodeBlocks, 256, 0, stream>>>(nidx, dsrc, ddst, alpha);
  k_rowcol <<<NN,         256, 0, stream>>>(z, nidx, alpha, convrow_w, convrow_b,
                                            convcol_w, convcol_b, outbuf, accum);
  k_bnparams<<<1, 32, 0, stream>>>(accum, bnrow_g, bnrow_b, bncol_g, bncol_b, params);
  k_bnapply<<<NN,         256, 0, stream>>>(outbuf, params);
  k_deg    <<<edgeBlocks, 256, 0, stream>>>(src, dst, dego, degi);
  k_featgc <<<tileBlocks, 256, 0, stream>>>(h, gc_w, dego, featgc);
  k_agg    <<<edgeBlocks, 256, 0, stream>>>(src, dst, featgc, agg);
  k_scores <<<nodeBlocks, 256, 0, stream>>>(agg, degi, gc_b, scores, partial);
  k_maxfin <<<1,          256, 0, stream>>>(partial, nodeBlocks, params);
  k_sumexp <<<nodeBlocks, 256, 0, stream>>>(scores, params);
  k_update <<<tileBlocks, 256, 0, stream>>>(outbuf, lwpad, h, h_proj_w,
                                            params, scores, hg);
  k_logits <<<1,          320, 0, stream>>>(hg, params, cls_w, cls_b, (float*)d_out);
}